// focal_loss_30494267802149
// MI455X (gfx1250) — compile-verified
//
#include <hip/hip_runtime.h>

// Focal loss, N=8.4M x C=4, f32. Memory-bound streaming reduction:
//   kernel 1: grid-stride accumulate -> WMMA wave reduce -> LDS -> ws[block]
//   kernel 2: reduce 2048 block partials -> * (1/N) -> d_out[0]
// Deterministic (no float atomics; fixed-order final sums).

typedef __attribute__((ext_vector_type(2))) float v2f;
typedef __attribute__((ext_vector_type(8))) float v8f;

#define WF 0.25f

// Per-sample focal loss term. Branchless gather over 4 classes.
__device__ __forceinline__ float sample_loss(const float4* __restrict__ logits,
                                             const int* __restrict__ lab32,
                                             long i) {
  float4 x = logits[i];           // global_load_b128, coalesced 512B/wave
  int lab = lab32[2 * i];         // low dword of little-endian int64 label
  float m = fmaxf(fmaxf(x.x, x.y), fmaxf(x.z, x.w));
  float e0 = __expf(x.x - m);
  float e1 = __expf(x.y - m);
  float e2 = __expf(x.z - m);
  float e3 = __expf(x.w - m);
  float S = (e0 + e1) + (e2 + e3);
  float xl = (lab == 0) ? x.x : (lab == 1) ? x.y : (lab == 2) ? x.z : x.w;
  float el = (lab == 0) ? e0  : (lab == 1) ? e1  : (lab == 2) ? e2  : e3;
  float w  = (lab == 0) ? (1.0f - WF) : (lab == 1) ? WF : 0.0f;
  float logp = (xl - m) - __logf(S);          // log p, always finite (p > 0)
  float p    = el * __builtin_amdgcn_rcpf(S); // p = e_l / S via v_rcp_f32
  float om   = 1.0f - p;
  return w * om * om * (-logp);               // -w (1-p)^2 log p
}

// Wave32 sum of 64 partials (acc0, acc1 per lane) via V_WMMA_F32_16X16X4_F32.
// A (16x4, documented layout): VGPR0 = acc0, VGPR1 = acc1; lanes 0-15 are
// K={0,1}, lanes 16-31 are K={2,3}. B = ones (layout-invariant), C = 0.
// D row m (replicated over columns) = acc0_m + acc1_m + acc0_{m+16} + acc1_{m+16}.
// Per-lane sum of the 8 D VGPRs = rows 0-7 (lanes<16) / rows 8-15 (lanes>=16);
// one shfl_xor(16) yields the full wave sum in every lane.
// Requires EXEC all-ones (guaranteed: called only at reconverged points).
__device__ __forceinline__ float wave_reduce_wmma(float a0, float a1) {
  v2f a; a[0] = a0;  a[1] = a1;
  v2f b; b[0] = 1.f; b[1] = 1.f;
  v8f c = {};
  c = __builtin_amdgcn_wmma_f32_16x16x4_f32(
      /*neg_a=*/false, a, /*neg_b=*/false, b,
      /*c_mod=*/(short)0, c, /*reuse_a=*/false, /*reuse_b=*/false);
  float s = 0.f;
#pragma unroll
  for (int k = 0; k < 8; ++k) s += c[k];
  s += __shfl_xor(s, 16, 32);
  return s;
}

__global__ void __launch_bounds__(256)
focal_partial_kernel(const float4* __restrict__ logits,
                     const int* __restrict__ lab32,
                     float* __restrict__ partial, long N) {
  long tid    = (long)blockIdx.x * blockDim.x + threadIdx.x;
  long stride = (long)gridDim.x * blockDim.x;
  float acc0 = 0.f, acc1 = 0.f;  // two chains -> ILP + WMMA A operand
  for (long i = tid; i < N; i += 2 * stride) {
    __builtin_prefetch((const void*)(logits + i + 2 * stride), 0, 0); // global_prefetch_b8
    acc0 += sample_loss(logits, lab32, i);
    long j = i + stride;
    if (j < N) acc1 += sample_loss(logits, lab32, j);
  }
  // Reconverged here; EXEC all-ones for WMMA.
  float s = wave_reduce_wmma(acc0, acc1);

  __shared__ float wsums[8];
  int lane = threadIdx.x & 31;
  int wv   = threadIdx.x >> 5;
  if (lane == 0) wsums[wv] = s;
  __syncthreads();
  if (threadIdx.x == 0) {
    float t = 0.f;
#pragma unroll
    for (int k = 0; k < 8; ++k) t += wsums[k];  // fixed order: deterministic
    partial[blockIdx.x] = t;
  }
}

__global__ void __launch_bounds__(256)
focal_final_kernel(const float* __restrict__ partial, int B,
                   float* __restrict__ out, float invN) {
  float acc0 = 0.f, acc1 = 0.f;
  for (int j = threadIdx.x; j < B; j += 512) {  // uniform trip count (B % 512 == 0)
    acc0 += partial[j];
    int k = j + 256;
    if (k < B) acc1 += partial[k];
  }
  float s = wave_reduce_wmma(acc0, acc1);

  __shared__ float wsums[8];
  int lane = threadIdx.x & 31;
  int wv   = threadIdx.x >> 5;
  if (lane == 0) wsums[wv] = s;
  __syncthreads();
  if (threadIdx.x == 0) {
    float t = 0.f;
#pragma unroll
    for (int k = 0; k < 8; ++k) t += wsums[k];
    out[0] = t * invN;
  }
}

extern "C" void kernel_launch(void* const* d_in, const int* in_sizes, int n_in,
                              void* d_out, int out_size, void* d_ws, size_t ws_size,
                              hipStream_t stream) {
  const float4* logits = (const float4*)d_in[0];   // [N,4] f32, 16B rows
  const int*    lab32  = (const int*)d_in[1];      // int64 labels, read low dword
  long N = in_sizes[1];

  float* partial = (float*)d_ws;                   // 2048 floats = 8 KB scratch
  const int BLOCKS = 2048;                         // 16K wave32 in flight -> HBM-saturating

  focal_partial_kernel<<<BLOCKS, 256, 0, stream>>>(logits, lab32, partial, N);
  float invN = (float)(1.0 / (double)N);
  focal_final_kernel<<<1, 256, 0, stream>>>(partial, BLOCKS, (float*)d_out, invN);
}